// LeaderScoreGNN_3994319585551
// MI455X (gfx1250) — compile-verified
//
#include <hip/hip_runtime.h>
#include <hip/hip_bf16.h>

typedef float v2f __attribute__((ext_vector_type(2)));
typedef float v8f __attribute__((ext_vector_type(8)));

#define F_IN 128
#define F_HID 64

// ---------------------------------------------------------------------------
// K0: init accumulators. deg starts at 1.0 (self-loop weight), degw at 0.
// ---------------------------------------------------------------------------
__global__ void k_init(float* __restrict__ deg, float* __restrict__ degw,
                       unsigned* __restrict__ maxslot, int N) {
  int i = blockIdx.x * blockDim.x + threadIdx.x;
  if (i < N) { deg[i] = 1.0f; degw[i] = 0.0f; }
  if (i == 0) *maxslot = 0u;
}

// ---------------------------------------------------------------------------
// K1: weighted in-degree (by dst) and weighted out-degree (by src, for score)
// ---------------------------------------------------------------------------
__global__ void k_deg(const long long* __restrict__ src, const long long* __restrict__ dst,
                      const float* __restrict__ w, float* __restrict__ deg,
                      float* __restrict__ degw, int E) {
  int e = blockIdx.x * blockDim.x + threadIdx.x;
  if (e >= E) return;
  float we = w[e];
  atomicAdd(&deg[dst[e]], we);
  atomicAdd(&degw[src[e]], we);
}

// ---------------------------------------------------------------------------
// K2: dinv = rsqrt(deg) (deg >= 1 always); fused wave-max of degw (>=0, so
// uint-ordered atomicMax is exact for floats).
// ---------------------------------------------------------------------------
__global__ void k_dinv_max(const float* __restrict__ deg, const float* __restrict__ degw,
                           float* __restrict__ dinv, unsigned* __restrict__ maxslot, int N) {
  int i = blockIdx.x * blockDim.x + threadIdx.x;
  float dw = 0.0f;
  if (i < N) { dinv[i] = rsqrtf(deg[i]); dw = degw[i]; }
  for (int off = 16; off; off >>= 1) dw = fmaxf(dw, __shfl_xor(dw, off, 32));
  if ((threadIdx.x & 31) == 0) atomicMax(maxslot, __float_as_uint(dw));
}

// ---------------------------------------------------------------------------
// K3: per-edge symmetric normalization
// ---------------------------------------------------------------------------
__global__ void k_norm(const long long* __restrict__ src, const long long* __restrict__ dst,
                       const float* __restrict__ w, const float* __restrict__ dinv,
                       float* __restrict__ norm, int E) {
  int e = blockIdx.x * blockDim.x + threadIdx.x;
  if (e >= E) return;
  norm[e] = dinv[src[e]] * w[e] * dinv[dst[e]];
}

// ---------------------------------------------------------------------------
// K4: h1 = x @ W1 via V_WMMA_F32_16X16X4_F32.
// W1 is staged global->LDS with GLOBAL_LOAD_ASYNC_TO_LDS_B128 (ASYNCcnt),
// then pair-swizzled in LDS so each B fragment (W1[k][n], W1[k+1][n]) is a
// contiguous 8-byte v2f -> single ds_load_b64, no re-pairing v_movs.
// Fragment layouts per ISA 7.12.2 (wave32):
//   A 16x4 f32: lanes 0-15 hold K={0,1}, lanes 16-31 hold K={2,3}, M=lane&15
//   B 4x16 f32: row-striped across lanes within VGPR (mirrors C/D halves)
//   C/D 16x16:  VGPR j -> M = j + 8*(lane>>4), N = lane&15
// ---------------------------------------------------------------------------
__global__ __launch_bounds__(256) void k_gemm_wmma(const float* __restrict__ x,
                                                   const float* __restrict__ W1,
                                                   float* __restrict__ h1, int N) {
  __shared__ float w1raw[F_IN * F_HID];  // async-copy landing zone (row-major)
  __shared__ float w1s[F_IN * F_HID];    // swizzled: [(k>>1)*128 + n*2 + (k&1)]
  const int tid = threadIdx.x;

  // --- async copy W1 (32 KB) into LDS: 16 B per lane-issue, ASYNCcnt ---
  {
    const unsigned lbase = (unsigned)(size_t)(const void*)w1raw; // low 32b of
    const unsigned long long gbase = (unsigned long long)(size_t)W1; // flat = LDS off
    #pragma unroll
    for (int i = 0; i < (F_IN * F_HID * 4) / (256 * 16); ++i) {
      const unsigned byteoff = (unsigned)(tid + i * 256) * 16u;
      const unsigned ldsa = lbase + byteoff;
      const unsigned long long ga = gbase + byteoff;
      asm volatile("global_load_async_to_lds_b128 %0, %1, off"
                   :: "v"(ldsa), "v"(ga) : "memory");
    }
#if __has_builtin(__builtin_amdgcn_s_wait_asynccnt)
    __builtin_amdgcn_s_wait_asynccnt(0);
#else
    asm volatile("s_wait_asynccnt 0" ::: "memory");
#endif
  }
  __syncthreads();

  // --- LDS->LDS pair swizzle ---
  for (int i = tid; i < F_IN * F_HID; i += 256) {
    const int k = i >> 6, n = i & 63;
    w1s[((k >> 1) << 7) + (n << 1) + (k & 1)] = w1raw[i];
  }
  __syncthreads();

  const int wave = tid >> 5;
  const int lane = tid & 31;
  const int g = lane >> 4;       // half-wave group
  const int l16 = lane & 15;
  const int mtile = blockIdx.x * 8 + wave;
  const int mtiles = (N + 15) >> 4;
  if (mtile >= mtiles) return;

  int row = mtile * 16 + l16;
  if (row >= N) row = N - 1;     // clamp (EXEC must stay all-1 for WMMA)
  const float* xrow = x + (size_t)row * F_IN;

  v8f acc0 = {}, acc1 = {}, acc2 = {}, acc3 = {};
  for (int kk = 0; kk < F_IN; kk += 4) {
    const int kb = kk + 2 * g;                    // kb even -> 8B aligned
    const v2f a = *(const v2f*)&xrow[kb];
    const float* bp = &w1s[((kb >> 1) << 7) + (l16 << 1)];
    const v2f b0 = *(const v2f*)(bp);
    const v2f b1 = *(const v2f*)(bp + 32);
    const v2f b2 = *(const v2f*)(bp + 64);
    const v2f b3 = *(const v2f*)(bp + 96);
    acc0 = __builtin_amdgcn_wmma_f32_16x16x4_f32(false, a, false, b0, (short)0, acc0, false, false);
    acc1 = __builtin_amdgcn_wmma_f32_16x16x4_f32(false, a, false, b1, (short)0, acc1, false, false);
    acc2 = __builtin_amdgcn_wmma_f32_16x16x4_f32(false, a, false, b2, (short)0, acc2, false, false);
    acc3 = __builtin_amdgcn_wmma_f32_16x16x4_f32(false, a, false, b3, (short)0, acc3, false, false);
  }

  #pragma unroll
  for (int j = 0; j < 8; ++j) {
    const int r = mtile * 16 + j + 8 * g;
    if (r < N) {
      float* o = h1 + (size_t)r * F_HID + l16;
      o[0]  = acc0[j];
      o[16] = acc1[j];
      o[32] = acc2[j];
      o[48] = acc3[j];
    }
  }
}

// ---------------------------------------------------------------------------
// K5: initialize agg1 with the self-loop term: agg1[i,f] = h1[i,f]*dinv[i]^2
// ---------------------------------------------------------------------------
__global__ void k_self1(const float* __restrict__ h1, const float* __restrict__ dinv,
                        float* __restrict__ agg1, int NF) {
  int t = blockIdx.x * blockDim.x + threadIdx.x;
  if (t >= NF) return;
  float di = dinv[t >> 6];
  agg1[t] = h1[t] * di * di;
}

// ---------------------------------------------------------------------------
// K6: edge scatter for layer 1. 64 consecutive threads per edge: coalesced
// gather of h1[src], cacheline-clustered atomic adds onto agg1[dst].
// ---------------------------------------------------------------------------
__global__ void k_agg1(const long long* __restrict__ src, const long long* __restrict__ dst,
                       const float* __restrict__ norm, const float* __restrict__ h1,
                       float* __restrict__ agg1, long long EF) {
  long long t = (long long)blockIdx.x * blockDim.x + threadIdx.x;
  if (t >= EF) return;
  long long e = t >> 6;
  int f = (int)(t & 63);
  long long s = src[e], d = dst[e];
  atomicAdd(&agg1[d * F_HID + f], h1[s * F_HID + f] * norm[e]);
}

// ---------------------------------------------------------------------------
// K7: wave-per-node: t[i] = relu(agg1[i,:]+b1) . W2  (64-wide dot, wave32
// shuffle tree); seed layer-2 accumulator with its self-loop term.
// ---------------------------------------------------------------------------
__global__ __launch_bounds__(256) void k_hidden(const float* __restrict__ agg1,
                                                const float* __restrict__ b1,
                                                const float* __restrict__ W2,
                                                const float* __restrict__ dinv,
                                                float* __restrict__ t, float* __restrict__ agg2,
                                                int N) {
  int node = (int)(((long long)blockIdx.x * blockDim.x + threadIdx.x) >> 5);
  int lane = threadIdx.x & 31;
  if (node >= N) return;
  const float* row = agg1 + (size_t)node * F_HID;
  float s = fmaxf(row[lane] + b1[lane], 0.0f) * W2[lane]
          + fmaxf(row[lane + 32] + b1[lane + 32], 0.0f) * W2[lane + 32];
  for (int off = 16; off; off >>= 1) s += __shfl_xor(s, off, 32);
  if (lane == 0) {
    t[node] = s;
    float di = dinv[node];
    agg2[node] = s * di * di;
  }
}

// ---------------------------------------------------------------------------
// K8: scalar edge scatter for layer 2
// ---------------------------------------------------------------------------
__global__ void k_agg2(const long long* __restrict__ src, const long long* __restrict__ dst,
                       const float* __restrict__ norm, const float* __restrict__ t,
                       float* __restrict__ agg2, int E) {
  int e = blockIdx.x * blockDim.x + threadIdx.x;
  if (e >= E) return;
  atomicAdd(&agg2[dst[e]], t[src[e]] * norm[e]);
}

// ---------------------------------------------------------------------------
// K9: epilogue: sigmoid + degree-weighted score
// ---------------------------------------------------------------------------
__global__ void k_final(const float* __restrict__ agg2, const float* __restrict__ b2,
                        const float* __restrict__ degw, const unsigned* __restrict__ maxslot,
                        float* __restrict__ out, int N) {
  int i = blockIdx.x * blockDim.x + threadIdx.x;
  if (i >= N) return;
  float z = agg2[i] + b2[0];
  float sc = 1.0f / (1.0f + expf(-z));
  float m = __uint_as_float(*maxslot);
  out[i] = sc * (1.0f + degw[i] / m);
}

// ---------------------------------------------------------------------------
extern "C" void kernel_launch(void* const* d_in, const int* in_sizes, int n_in,
                              void* d_out, int out_size, void* d_ws, size_t ws_size,
                              hipStream_t stream) {
  const float*     x   = (const float*)d_in[0];
  const long long* ei  = (const long long*)d_in[1];   // int64 [2, E]
  const float*     ew  = (const float*)d_in[2];
  const float*     W1  = (const float*)d_in[3];
  const float*     b1  = (const float*)d_in[4];
  const float*     W2  = (const float*)d_in[5];
  const float*     b2  = (const float*)d_in[6];
  float*           out = (float*)d_out;

  const int N = in_sizes[0] / F_IN;
  const int E = in_sizes[1] / 2;
  const long long* src = ei;
  const long long* dst = ei + E;

  // workspace carve (floats), 256-element aligned slices
  float* ws = (float*)d_ws;
  size_t off = 0;
  auto take = [&](size_t n) { float* p = ws + off; off += (n + 255) & ~(size_t)255; return p; };
  float*    deg     = take(N);
  float*    dinv    = take(N);
  float*    degw    = take(N);
  float*    tbuf    = take(N);
  float*    agg2    = take(N);
  unsigned* maxslot = (unsigned*)take(256);
  float*    norm    = take(E);
  float*    h1      = take((size_t)N * F_HID);
  float*    agg1    = take((size_t)N * F_HID);
  (void)ws_size;

  const int B = 256;
  const int gN  = (N + B - 1) / B;
  const int gE  = (E + B - 1) / B;
  const int gNF = ((N * F_HID) + B - 1) / B;
  const long long EF = (long long)E * F_HID;
  const unsigned gEF = (unsigned)((EF + B - 1) / B);
  const int mtiles = (N + 15) / 16;
  const int gGemm = (mtiles + 7) / 8;         // 8 waves per block, 1 m-tile each
  const int gWav = (N * 32 + B - 1) / B;      // wave-per-node

  k_init     <<<gN, B, 0, stream>>>(deg, degw, maxslot, N);
  k_deg      <<<gE, B, 0, stream>>>(src, dst, ew, deg, degw, E);
  k_dinv_max <<<gN, B, 0, stream>>>(deg, degw, dinv, maxslot, N);
  k_norm     <<<gE, B, 0, stream>>>(src, dst, ew, dinv, norm, E);
  k_gemm_wmma<<<gGemm, B, 0, stream>>>(x, W1, h1, N);
  k_self1    <<<gNF, B, 0, stream>>>(h1, dinv, agg1, N * F_HID);
  k_agg1     <<<gEF, B, 0, stream>>>(src, dst, norm, h1, agg1, EF);
  k_hidden   <<<gWav, B, 0, stream>>>(agg1, b1, W2, dinv, tbuf, agg2, N);
  k_agg2     <<<gE, B, 0, stream>>>(src, dst, norm, tbuf, agg2, E);
  k_final    <<<gN, B, 0, stream>>>(agg2, b2, degw, maxslot, out, N);
}